// GCN_4973572129388
// MI455X (gfx1250) — compile-verified
//
#include <hip/hip_runtime.h>

// GCN forward on MI455X (gfx1250, wave32).
//   h   = relu( SpMM(A, x@W1) + b1 )
//   out =       SpMM(A, h@W2) + b2
// GEMMs: V_WMMA_F32_16X16X4_F32 (exact f32), B staged through LDS (transposed,
// padded) so each WMMA's B operand is one aligned ds_load_b64 with immediate
// offsets. SpMM is the runtime bottleneck; its dense operands (102MB / 51MB)
// are L2-resident (192MB L2); scatter via global_atomic_add_f32.

typedef __attribute__((ext_vector_type(2))) float v2f;
typedef __attribute__((ext_vector_type(8))) float v8f;

#define NNODE 50000
#define NFEAT 512
#define NHID  512
#define NOUT  256

// out[i] = bias[i & fmask]   (F is a power of two: 512 or 256)
__global__ void init_bias_kernel(float* __restrict__ out,
                                 const float* __restrict__ bias,
                                 long total, int fmask) {
  long i = (long)blockIdx.x * blockDim.x + threadIdx.x;
  if (i < total) out[i] = bias[(int)(i & (long)fmask)];
}

// C[M,Nc] = op(A)[M,K] @ B[K,Nc], op = identity or ReLU (applied to A loads).
// Block = 256 threads = 8 waves. Block owns a 16-row M-tile; wave w owns the
// N-strip [w*NT*16, (w+1)*NT*16). Nc = 8 * NT * 16.
// B is staged per 32-deep K-chunk into LDS as Bs[n][kk] (transposed, stride
// KC+2 floats => even, so (kk + 2*half) float2 loads are 8B-aligned).
// WMMA f32 16x16x4 VGPR layouts (ISA 7.12.2):
//   A: lane = 16*half + m, vgpr v -> A[m, k + 2*half + v]
//   B: lane = 16*half + n, vgpr v -> B[k + 2*half + v, n]
//   C/D: lane = 16*half + n, vgpr i -> C[i + 8*half, n]
template <int NT, bool RELU>
__global__ void gemm_wmma_lds(const float* __restrict__ A,
                              const float* __restrict__ B,
                              float* __restrict__ C,
                              int M, int K) {
  constexpr int Nc   = NT * 16 * 8;   // total columns (512 or 256)
  constexpr int KC   = 32;            // K-chunk staged in LDS
  constexpr int KSTR = KC + 2;        // padded (even) LDS row stride in floats
  __shared__ float Bs[Nc * KSTR];

  const int tid  = threadIdx.x;
  const int wv   = tid >> 5;
  const int lane = tid & 31;
  const int half = lane >> 4;         // 0: lanes 0-15, 1: lanes 16-31
  const int l15  = lane & 15;

  const int tm  = blockIdx.x << 4;    // 16 rows per block
  const int tn0 = wv * (NT * 16);     // this wave's column base

  v8f acc[NT];
#pragma unroll
  for (int t = 0; t < NT; ++t) acc[t] = (v8f){};

  const float* arow = A + (size_t)(tm + l15) * K + 2 * half;

  for (int k0 = 0; k0 < K; k0 += KC) {
    __syncthreads();  // previous chunk's Bs reads done before overwrite
    // Stage B[k0:k0+KC, 0:Nc] -> Bs[n][kk] (coalesced global reads)
    for (int i = tid; i < KC * Nc; i += 256) {
      int kk = i >> (NT == 4 ? 9 : 8);   // i / Nc
      int n  = i & (Nc - 1);
      Bs[n * KSTR + kk] = B[(size_t)(k0 + kk) * Nc + n];
    }
    __syncthreads();

#pragma unroll
    for (int kk = 0; kk < KC; kk += 4) {
      v2f a;
      float a0 = arow[k0 + kk];
      float a1 = arow[k0 + kk + 1];
      if (RELU) { a0 = fmaxf(a0, 0.0f); a1 = fmaxf(a1, 0.0f); }
      a[0] = a0; a[1] = a1;
#pragma unroll
      for (int t = 0; t < NT; ++t) {
        int n = tn0 + t * 16 + l15;
        v2f b = *(const v2f*)&Bs[n * KSTR + kk + 2 * half];
        acc[t] = __builtin_amdgcn_wmma_f32_16x16x4_f32(
            /*neg_a=*/false, a, /*neg_b=*/false, b,
            /*c_mod=*/(short)0, acc[t], /*reuse_a=*/false, /*reuse_b=*/false);
      }
    }
  }

#pragma unroll
  for (int i = 0; i < 8; ++i) {
    float* crow = C + (size_t)(tm + i + 8 * half) * Nc;
#pragma unroll
    for (int t = 0; t < NT; ++t) crow[tn0 + t * 16 + l15] = acc[t][i];
  }
}

// out[row[e], :] += val[e] * dense[col[e], :]   (one wave per edge)
// dense is L2-resident; float4 gathers, f32 atomic scatter.
__global__ void spmm_scatter_kernel(const int* __restrict__ erow,
                                    const int* __restrict__ ecol,
                                    const float* __restrict__ eval,
                                    const float* __restrict__ dense,
                                    float* __restrict__ out,
                                    int E, int F) {
  int gid  = blockIdx.x * blockDim.x + threadIdx.x;
  int wave = gid >> 5;
  int lane = threadIdx.x & 31;
  if (wave >= E) return;

  int   r = erow[wave];
  int   c = ecol[wave];
  float v = eval[wave];

  const float4* src = (const float4*)(dense + (size_t)c * F);
  float*        dst = out + (size_t)r * F;

  int nvec = F >> 2;
  for (int j = lane; j < nvec; j += 32) {
    float4 d = src[j];
    int base = j << 2;
    atomicAdd(dst + base + 0, v * d.x);
    atomicAdd(dst + base + 1, v * d.y);
    atomicAdd(dst + base + 2, v * d.z);
    atomicAdd(dst + base + 3, v * d.w);
  }
}

extern "C" void kernel_launch(void* const* d_in, const int* in_sizes, int n_in,
                              void* d_out, int out_size, void* d_ws, size_t ws_size,
                              hipStream_t stream) {
  const float* x    = (const float*)d_in[0];
  const int*   erow = (const int*)  d_in[1];
  const int*   ecol = (const int*)  d_in[2];
  const float* eval = (const float*)d_in[3];
  const float* W1   = (const float*)d_in[4];
  const float* b1   = (const float*)d_in[5];
  const float* W2   = (const float*)d_in[6];
  const float* b2   = (const float*)d_in[7];
  float*       out  = (float*)d_out;

  const int E = in_sizes[1];

  // Workspace layout (floats): s1[50000*512] | h[50000*512]; s2 reuses s1.
  float* s1 = (float*)d_ws;
  float* h  = s1 + (size_t)NNODE * NHID;
  float* s2 = s1;  // s1 dead after SpMM1

  const int BLK = 256;  // 8 waves / block

  // 1) h = b1 (broadcast)
  {
    long total = (long)NNODE * NHID;
    int blocks = (int)((total + BLK - 1) / BLK);
    init_bias_kernel<<<blocks, BLK, 0, stream>>>(h, b1, total, NHID - 1);
  }

  // 2) s1 = x @ W1   (Nc = 512 -> NT = 4)
  {
    int blocks = NNODE >> 4;  // 3125
    gemm_wmma_lds<4, false><<<blocks, BLK, 0, stream>>>(x, W1, s1,
                                                        NNODE, NFEAT);
  }

  // 3) h += SpMM(adj, s1)
  {
    int blocks = (E * 32 + BLK - 1) / BLK;
    spmm_scatter_kernel<<<blocks, BLK, 0, stream>>>(erow, ecol, eval, s1, h,
                                                    E, NHID);
  }

  // 4) s2 = relu(h) @ W2   (Nc = 256 -> NT = 2)
  {
    int blocks = NNODE >> 4;  // 3125
    gemm_wmma_lds<2, true><<<blocks, BLK, 0, stream>>>(h, W2, s2,
                                                       NNODE, NHID);
  }

  // 5) out = b2 (broadcast)
  {
    long total = (long)NNODE * NOUT;
    int blocks = (int)((total + BLK - 1) / BLK);
    init_bias_kernel<<<blocks, BLK, 0, stream>>>(out, b2, total, NOUT - 1);
  }

  // 6) out += SpMM(adj, s2)
  {
    int blocks = (E * 32 + BLK - 1) / BLK;
    spmm_scatter_kernel<<<blocks, BLK, 0, stream>>>(erow, ecol, eval, s2, out,
                                                    E, NOUT);
  }
}